// BertBltEmbeddings_80891414053419
// MI455X (gfx1250) — compile-verified
//
#include <hip/hip_runtime.h>
#include <stdint.h>

// BLT byte/pos/hash-ngram embedding + LayerNorm for MI455X (gfx1250, wave32).
// One block per sequence position s; one wave per batch row b.
// - pos_emb[s] staged to LDS once per block via CDNA5 async global->LDS copy.
// - Hash-table gathers use non-temporal loads (1.8 GB table, ~zero reuse) so
//   the 192 MB L2 stays reserved for byte_emb/pos_emb/gamma/beta.
// - Output stores are non-temporal (write-once stream).
// - Wave-uniform hash chain + row addresses scalarized via readlane.

typedef float f4 __attribute__((ext_vector_type(4)));

#define NGRAMS 6
static constexpr unsigned kHashVocab = 100000u;
static constexpr unsigned kBase      = 257u;
static constexpr float    kLnEps     = 1e-12f;
static constexpr int      kWaves     = 8;
static constexpr int      kBlock     = kWaves * 32;   // 256 threads = 8 wave32

__device__ __forceinline__ float wave_sum32(float v) {
#pragma unroll
    for (int o = 16; o > 0; o >>= 1) v += __shfl_xor(v, o, 32);
    return v;
}

// ---------------- specialized register-resident kernel (H = CHUNKS*128) ----
template <int CHUNKS>
__global__ void __launch_bounds__(kBlock)
blt_reg_kernel(const int*   __restrict__ ids,        // [B,S]
               const float* __restrict__ byte_emb,   // [260,H]
               const float* __restrict__ pos_emb,    // [S,H]
               const float* __restrict__ hash_tab,   // [6,100000,H]
               const float* __restrict__ ln_gamma,   // [H]
               const float* __restrict__ ln_beta,    // [H]
               float*       __restrict__ out,        // [B,S,H]
               int B, int S)
{
    constexpr int H = CHUNKS * 128;
    __shared__ float posRow[H];

    const int tid  = (int)threadIdx.x;
    const int lane = tid & 31;
    const int wave = tid >> 5;
    const int s    = (int)blockIdx.x;

    // ---- Async global->LDS stage of pos_emb[s][:] (ASYNCcnt path) ----
    {
        const float* src  = pos_emb + (size_t)s * (size_t)H;
        constexpr int nvec = H / 4;                     // 16B vectors
        for (int v = tid; v < nvec; v += kBlock) {
            uint32_t ldsAddr = (uint32_t)(uintptr_t)(&posRow[v * 4]);
            uint32_t goff    = (uint32_t)(v * 16);
            asm volatile("global_load_async_to_lds_b128 %0, %1, %2 offset:0"
                         :: "v"(ldsAddr), "v"(goff), "s"(src)
                         : "memory");
        }
    }
    asm volatile("s_wait_asynccnt 0" ::: "memory");
    __syncthreads();

    for (int b = wave; b < B; b += kWaves) {
        const int* idRow = ids + (size_t)b * (size_t)S;

        // lanes 0..7 fetch the id window ids[s-7..s] (0-padded at row start)
        int myid = 0;
        {
            int p = s - 7 + lane;
            if (lane < 8 && p >= 0) myid = idRow[p];
        }
        // scalarize the window: wk[k] lives in SGPRs
        unsigned wk[8];
#pragma unroll
        for (int k = 0; k < 8; ++k)
            wk[k] = (unsigned)__builtin_amdgcn_readlane(myid, k);

        // shared Horner chain (SALU): h_j = (h_{j-1}*257 + ids[s-j]) % 100000
        unsigned h = wk[7];
        unsigned hv[NGRAMS];
#pragma unroll
        for (int j = 1; j <= 7; ++j) {
            h = (h * kBase + wk[7 - j]) % kHashVocab;
            if (j >= 2) hv[j - 2] = h;
        }
#pragma unroll
        for (int t = 0; t < NGRAMS; ++t)
            if (s < t + 2) hv[t] = wk[7];     // n = t+3: s < n-1 -> ids[s]

        const f4* brow = (const f4*)(byte_emb + (size_t)wk[7] * (size_t)H);
        const f4* hrow[NGRAMS];
#pragma unroll
        for (int t = 0; t < NGRAMS; ++t)
            hrow[t] = (const f4*)(hash_tab +
                      ((size_t)t * (size_t)kHashVocab + (size_t)hv[t]) * (size_t)H);

        // Pass 1: gather-accumulate into registers, collect LN statistics
        f4 x[CHUNKS];
        float sum = 0.f, sumsq = 0.f;
        const float inv6 = 1.0f / (float)NGRAMS;
#pragma unroll
        for (int c = 0; c < CHUNKS; ++c) {
            const int ev = c * 32 + lane;               // f4 index within row
            f4 a  = brow[ev];                           // cached (hot, 260 rows)
            f4 pz = *(const f4*)(&posRow[ev * 4]);      // LDS
            f4 hs = __builtin_nontemporal_load(&hrow[0][ev]);
#pragma unroll
            for (int t = 1; t < NGRAMS; ++t)
                hs += __builtin_nontemporal_load(&hrow[t][ev]);
            f4 xv = a + pz + hs * inv6;
            x[c] = xv;
            sum   += xv.x + xv.y + xv.z + xv.w;
            sumsq += xv.x * xv.x + xv.y * xv.y + xv.z * xv.z + xv.w * xv.w;
        }

        sum   = wave_sum32(sum);
        sumsq = wave_sum32(sumsq);
        const float invH = 1.0f / (float)H;
        const float mu   = sum * invH;
        float var = sumsq * invH - mu * mu;
        var = var < 0.f ? 0.f : var;
        const float rstd = rsqrtf(var + kLnEps);

        // Pass 2: normalize + affine, streaming NT stores
        f4* orow = (f4*)(out + ((size_t)b * (size_t)S + (size_t)s) * (size_t)H);
#pragma unroll
        for (int c = 0; c < CHUNKS; ++c) {
            const int ev = c * 32 + lane;
            f4 g  = *(const f4*)(ln_gamma + ev * 4);
            f4 be = *(const f4*)(ln_beta  + ev * 4);
            f4 o  = (x[c] - mu) * rstd * g + be;
            __builtin_nontemporal_store(o, &orow[ev]);
        }
    }
}

// ---------------- generic fallback (any H % 4 == 0), LDS x-staging ---------
__global__ void __launch_bounds__(kBlock)
blt_gen_kernel(const int*   __restrict__ ids,
               const float* __restrict__ byte_emb,
               const float* __restrict__ pos_emb,
               const float* __restrict__ hash_tab,
               const float* __restrict__ ln_gamma,
               const float* __restrict__ ln_beta,
               float*       __restrict__ out,
               int B, int S, int H)
{
    extern __shared__ float lds[];
    float* posRow = lds;
    float* xbuf   = lds + H;

    const int tid  = (int)threadIdx.x;
    const int lane = tid & 31;
    const int wave = tid >> 5;
    const int s    = (int)blockIdx.x;

    for (int e = tid; e * 4 < H; e += kBlock)
        *(f4*)(posRow + e * 4) = *(const f4*)(pos_emb + (size_t)s * H + e * 4);
    __syncthreads();

    for (int b = wave; b < B; b += kWaves) {
        const int* idRow = ids + (size_t)b * (size_t)S;
        int myid = 0;
        {
            int p = s - 7 + lane;
            if (lane < 8 && p >= 0) myid = idRow[p];
        }
        unsigned wk[8];
#pragma unroll
        for (int k = 0; k < 8; ++k)
            wk[k] = (unsigned)__builtin_amdgcn_readlane(myid, k);

        unsigned h = wk[7];
        unsigned hv[NGRAMS];
#pragma unroll
        for (int j = 1; j <= 7; ++j) {
            h = (h * kBase + wk[7 - j]) % kHashVocab;
            if (j >= 2) hv[j - 2] = h;
        }
#pragma unroll
        for (int t = 0; t < NGRAMS; ++t)
            if (s < t + 2) hv[t] = wk[7];

        const float* brow = byte_emb + (size_t)wk[7] * (size_t)H;
        const float* hrow[NGRAMS];
#pragma unroll
        for (int t = 0; t < NGRAMS; ++t)
            hrow[t] = hash_tab + ((size_t)t * (size_t)kHashVocab + (size_t)hv[t]) * (size_t)H;

        float* xw = xbuf + (size_t)wave * (size_t)H;
        float sum = 0.f, sumsq = 0.f;
        const float inv6 = 1.0f / (float)NGRAMS;
        for (int e4 = lane; e4 * 4 < H; e4 += 32) {
            const int e = e4 * 4;
            f4 a  = *(const f4*)(brow + e);
            f4 pz = *(const f4*)(posRow + e);
            f4 hs = __builtin_nontemporal_load((const f4*)(hrow[0] + e));
#pragma unroll
            for (int t = 1; t < NGRAMS; ++t)
                hs += __builtin_nontemporal_load((const f4*)(hrow[t] + e));
            f4 xv = a + pz + hs * inv6;
            *(f4*)(xw + e) = xv;
            sum   += xv.x + xv.y + xv.z + xv.w;
            sumsq += xv.x * xv.x + xv.y * xv.y + xv.z * xv.z + xv.w * xv.w;
        }
        sum   = wave_sum32(sum);
        sumsq = wave_sum32(sumsq);
        const float invH = 1.0f / (float)H;
        const float mu   = sum * invH;
        float var = sumsq * invH - mu * mu;
        var = var < 0.f ? 0.f : var;
        const float rstd = rsqrtf(var + kLnEps);

        float* orow = out + ((size_t)b * (size_t)S + (size_t)s) * (size_t)H;
        for (int e4 = lane; e4 * 4 < H; e4 += 32) {
            const int e = e4 * 4;
            f4 xv = *(const f4*)(xw + e);
            f4 g  = *(const f4*)(ln_gamma + e);
            f4 be = *(const f4*)(ln_beta  + e);
            f4 o  = (xv - mu) * rstd * g + be;
            __builtin_nontemporal_store(o, (f4*)(orow + e));
        }
    }
}

extern "C" void kernel_launch(void* const* d_in, const int* in_sizes, int n_in,
                              void* d_out, int out_size, void* d_ws, size_t ws_size,
                              hipStream_t stream)
{
    const int*   ids      = (const int*)d_in[0];
    const float* byte_emb = (const float*)d_in[1];
    const float* pos_emb  = (const float*)d_in[2];
    const float* hash_tab = (const float*)d_in[3];
    const float* ln_gamma = (const float*)d_in[4];
    const float* ln_beta  = (const float*)d_in[5];
    float*       out      = (float*)d_out;

    const int H = in_sizes[4];            // 768
    const int S = in_sizes[2] / H;        // 4096
    const int B = in_sizes[0] / S;        // 8

    const int chunks = (H % 128 == 0) ? (H / 128) : 0;
    switch (chunks) {
    case 4:
        blt_reg_kernel<4><<<S, kBlock, 0, stream>>>(
            ids, byte_emb, pos_emb, hash_tab, ln_gamma, ln_beta, out, B, S);
        break;
    case 6:   // H = 768 (the harness shape)
        blt_reg_kernel<6><<<S, kBlock, 0, stream>>>(
            ids, byte_emb, pos_emb, hash_tab, ln_gamma, ln_beta, out, B, S);
        break;
    case 8:
        blt_reg_kernel<8><<<S, kBlock, 0, stream>>>(
            ids, byte_emb, pos_emb, hash_tab, ln_gamma, ln_beta, out, B, S);
        break;
    default: {
        const size_t shmem = (size_t)(1 + kWaves) * (size_t)H * sizeof(float);
        blt_gen_kernel<<<S, kBlock, shmem, stream>>>(
            ids, byte_emb, pos_emb, hash_tab, ln_gamma, ln_beta, out, B, S, H);
        break;
    }
    }
}